// YOLOLoss_85916525789213
// MI455X (gfx1250) — compile-verified
//
#include <hip/hip_runtime.h>

// YOLO loss for MI455X (gfx1250, wave32).
// Sparse strategy: only the conf channel (1.2 MB) + <=800 obj cells of the
// 104 MB pred tensor are read. Workspace holds a per-cell winning-target
// table. Per block: celltab+conf chunks are staged to LDS with the CDNA5
// async-to-LDS path (ASYNCcnt), and the wave-level reduction runs through
// the f32 WMMA (B = ones => D columns replicate the wave row-sums).

typedef __attribute__((ext_vector_type(2))) float v2f;
typedef __attribute__((ext_vector_type(8))) float v8f;

#define NB    16
#define NT    50
#define NCLS  80
#define NA    3
#define GRD   80
#define CELLS (GRD*GRD)          // 6400
#define NCH   (5+NCLS)           // 85
#define TOTCH (NA*NCH)           // 255
#define NCELL (NB*NA*CELLS)      // 307200
#define SPLIT 8                  // chunks per (b, anchor) plane
#define CHUNK (CELLS/SPLIT)      // 800 cells per block
#define NBLK  (NB*NA*SPLIT)      // 384 blocks

__device__ __forceinline__ float sp_(float z) {          // stable softplus
  return fmaxf(z, 0.0f) + log1pf(expf(-fabsf(z)));
}
__device__ __forceinline__ float sg_(float z) { return 1.0f/(1.0f+expf(-z)); }
__device__ __forceinline__ float bce_(float z, float t) { // bce-with-logits
  return t*sp_(-z) + (1.0f - t)*sp_(z);
}

// Async global->LDS copy of one dword per lane (CDNA5 GLOBAL_LOAD_ASYNC_TO_LDS_B32).
// For generic pointers into LDS space, addr[31:0] is the LDS byte offset
// (ISA 10.2), which is exactly what the instruction's LDS-address VGPR takes.
__device__ __forceinline__ void async_copy_b32(void* lds_dst, const void* gsrc) {
  unsigned int  loff = (unsigned int)(unsigned long long)lds_dst;
  unsigned long long ga = (unsigned long long)gsrc;
  asm volatile("global_load_async_to_lds_b32 %0, %1, off"
               :: "v"(loff), "v"(ga) : "memory");
}
__device__ __forceinline__ void async_wait0() {
  asm volatile("s_wait_asynccnt 0" ::: "memory");
}

__global__ void yolo_init(int* __restrict__ celltab, float* __restrict__ out) {
  int i = blockIdx.x*blockDim.x + threadIdx.x;
  if (i < NCELL) celltab[i] = -1;
  if (i == 0) out[0] = 0.0f;
}

__global__ void yolo_scatter(const float* __restrict__ targets,
                             int* __restrict__ celltab) {
  int t = blockIdx.x*blockDim.x + threadIdx.x;
  if (t >= NB*NT) return;
  const float* row = targets + (size_t)t*5;
  float cls = row[0], fx = row[1], fy = row[2], fw = row[3], fh = row[4];
  if (cls + fx + fy + fw + fh == 0.0f) return;           // valid: sum != 0
  float gx = fx*GRD, gy = fy*GRD, gw = fw*GRD, gh = fh*GRD;
  int gi = (int)gx, gj = (int)gy;
  if (gi >= GRD || gj >= GRD || gi < 0 || gj < 0) return;
  const float AW[3] = {14.5f, 19.5f, 46.625f};           // anchors / stride(8)
  const float AH[3] = {11.25f, 24.75f, 40.75f};
  float ga = gw*gh;
  int best = 0; float br = -1.0f;
  #pragma unroll
  for (int n = 0; n < 3; n++) {                          // argmax, first-wins
    float inter = fminf(gw, AW[n]) * fminf(gh, AH[n]);
    float r = inter / (ga + AW[n]*AH[n] - inter + 1e-6f);
    if (r > br) { br = r; best = n; }
  }
  int b = t / NT;
  int idx = (b*NA + best)*CELLS + gj*GRD + gi;
  atomicMax(&celltab[idx], t);                           // last target wins
}

__global__ __launch_bounds__(256) void yolo_loss(
    const float* __restrict__ pred, const float* __restrict__ targets,
    const int* __restrict__ celltab, float* __restrict__ out) {
  __shared__ int   s_tab[CHUNK];
  __shared__ float s_conf[CHUNK];
  __shared__ float wsum[8];

  int ba    = blockIdx.x / SPLIT;                        // (b, anchor) plane
  int chunk = blockIdx.x % SPLIT;
  int b = ba / NA, a = ba % NA;
  const float* base = pred + ((size_t)b*TOTCH + (size_t)a*NCH) * CELLS;
  const float AW[3] = {14.5f, 19.5f, 46.625f};
  const float AH[3] = {11.25f, 24.75f, 40.75f};
  float aw = AW[a], ah = AH[a];
  const int tid   = threadIdx.x;
  const int cbase = chunk*CHUNK;                         // first cell of chunk
  const int* tabg = celltab + (size_t)ba*CELLS + cbase;
  const float* confg = base + 4*CELLS + cbase;

  // Stage celltab + conf chunks into LDS via the async path (ASYNCcnt).
  for (int i = tid; i < CHUNK; i += 256) {
    async_copy_b32(&s_tab[i],  &tabg[i]);
    async_copy_b32(&s_conf[i], &confg[i]);
  }
  async_wait0();          // this wave's transfers complete
  __syncthreads();        // all waves' transfers visible

  float acc = 0.0f;
  for (int i = tid; i < CHUNK; i += 256) {
    int   twin = s_tab[i];
    float zc   = s_conf[i];                              // conf logit
    int   cg   = cbase + i;                              // cell in plane
    float contrib;
    if (twin < 0) {                                      // noobj cell
      contrib = 0.5f * sp_(sg_(zc));                     // lambda_noobj folded
    } else {                                             // obj cell (rare)
      const float* row = targets + (size_t)twin*5;
      int   cid = (int)row[0];
      float gx = row[1]*GRD, gy = row[2]*GRD, gw = row[3]*GRD, gh = row[4]*GRD;
      float tx = gx - (float)(cg % GRD);
      float ty = gy - (float)(cg / GRD);
      float twv = logf(gw/aw + 1e-16f);
      float thv = logf(gh/ah + 1e-16f);
      float p0 = base[0*CELLS + cg], p1 = base[1*CELLS + cg];
      float p2 = base[2*CELLS + cg], p3 = base[3*CELLS + cg];
      float lx = bce_(sg_(p0), tx);
      float ly = bce_(sg_(p1), ty);
      float lw = (p2 - twv)*(p2 - twv);
      float lh = (p3 - thv)*(p3 - thv);
      float lco = sp_(-sg_(zc));                         // bce(sigmoid(z), 1)
      float lcls = 0.0f;
      for (int k = 0; k < NCLS; k++) {
        float z = base[(size_t)(5+k)*CELLS + cg];
        lcls += (k == cid) ? sp_(-z) : sp_(z);
      }
      contrib = 5.0f*(lx+ly+lw+lh) + lco + lcls;         // lambda_coord folded
    }
    acc += contrib;
  }

  // In-wave reduction through f32 WMMA: D = A x ones(+0). EXEC is all-ones
  // here (reconverged structured control flow). Every column of D carries the
  // 16 row-sums, so sum(all D elements) == 16 * (wave total).
  v2f Av; Av.x = acc;  Av.y = 0.0f;
  v2f Bv; Bv.x = 1.0f; Bv.y = 1.0f;
  v8f Cv = {};
  v8f D = __builtin_amdgcn_wmma_f32_16x16x4_f32(
      /*neg_a=*/false, Av, /*neg_b=*/false, Bv,
      /*c_mod=*/(short)0, Cv, /*reuse_a=*/false, /*reuse_b=*/false);
  float s = D[0]+D[1]+D[2]+D[3]+D[4]+D[5]+D[6]+D[7];
  for (int off = 16; off > 0; off >>= 1) s += __shfl_xor(s, off, 32);
  int wid = tid >> 5, lane = tid & 31;
  if (lane == 0) wsum[wid] = s;
  __syncthreads();
  if (tid == 0) {
    float tot = 0.0f;
    #pragma unroll
    for (int w = 0; w < 8; w++) tot += wsum[w];
    tot *= (1.0f/16.0f);   // undo WMMA 16-column replication
    tot *= (1.0f/16.0f);   // divide by batch size Bn
    atomicAdd(out, tot);
  }
}

extern "C" void kernel_launch(void* const* d_in, const int* in_sizes, int n_in,
                              void* d_out, int out_size, void* d_ws, size_t ws_size,
                              hipStream_t stream) {
  const float* pred    = (const float*)d_in[0];   // (16, 255, 80, 80) f32
  const float* targets = (const float*)d_in[1];   // (16, 50, 5) f32
  float* out = (float*)d_out;                     // scalar
  int* celltab = (int*)d_ws;                      // NCELL ints = 1.2 MB scratch
  yolo_init   <<<(NCELL + 255)/256, 256, 0, stream>>>(celltab, out);
  yolo_scatter<<<(NB*NT + 255)/256, 256, 0, stream>>>(targets, celltab);
  yolo_loss   <<<NBLK, 256, 0, stream>>>(pred, targets, celltab, out);
}